// mass_preserving_advection_62663572849366
// MI455X (gfx1250) — compile-verified
//
#include <hip/hip_runtime.h>

// Problem constants (fixed by setup_inputs): B=8, C=3, H=W=1024, fp32.
#define B_  8
#define C_  3
#define H_  1024
#define W_  1024
#define NPLANES (B_ * C_)        // 24
#define PLANE   (H_ * W_)        // 1048576

#define TILE     64              // spatial tile owned by one workgroup
#define HALO     8               // 8-sigma halo for N(0,1) displacements
#define LTW      (TILE + 2*HALO) // 80 : LDS tile width/height
#define LSTRIDE  (LTW + 1)       // 81 : odd stride -> decorrelated LDS banks
#define NTHREADS 256             // 8 wave32 per workgroup

typedef float v4f __attribute__((ext_vector_type(4)));

// ---------------------------------------------------------------------------
// Zero-init the output (d_out is poisoned before timing; must be deterministic
// every call). Vectorized 16B stores.
// ---------------------------------------------------------------------------
__global__ __launch_bounds__(256) void zero_f4(v4f* __restrict__ out, int n4) {
    int i = blockIdx.x * 256 + threadIdx.x;
    if (i < n4) {
        v4f z = {0.f, 0.f, 0.f, 0.f};
        out[i] = z;
    }
}

// ---------------------------------------------------------------------------
// Scatter advection with LDS privatization.
// One workgroup = one 64x64 tile of one (b,c) plane.
// Inputs streamed with non-temporal b128 loads (keep output hot in L2).
// Interior bilinear scatter -> ds_add_f32 into an 80x80 LDS tile (+halo),
// rare out-of-halo targets -> direct global_atomic_add_f32.
// Cooperative flush adds the LDS tile into global memory, skipping zeros.
// ---------------------------------------------------------------------------
__global__ __launch_bounds__(NTHREADS) void advect_scatter(
    const float* __restrict__ img,
    const float* __restrict__ U,
    const float* __restrict__ V,
    float* __restrict__ out)
{
    __shared__ float tile[LTW * LSTRIDE];

    // Zero the LDS accumulator tile.
    for (int i = threadIdx.x; i < LTW * LSTRIDE; i += NTHREADS)
        tile[i] = 0.0f;
    __syncthreads();

    const int tilesX = W_ / TILE;                 // 16
    const int tilesY = H_ / TILE;                 // 16
    const int bid    = blockIdx.x;
    const int tX     = bid % tilesX;
    const int tY     = (bid / tilesX) % tilesY;
    const int plane  = bid / (tilesX * tilesY);   // 0 .. 23
    const int tx0    = tX * TILE;
    const int ty0    = tY * TILE;
    const long long base = (long long)plane * PLANE;

    // Layout: 16 threads cover one row (4 contiguous cols each = 64 cols),
    // 256 threads cover 16 rows per iteration; 4 iterations = 64 rows.
    const int c4  = (threadIdx.x & 15) << 2;      // column group start 0..60
    const int r0  = threadIdx.x >> 4;             // 0..15 (row in group)

    for (int it = 0; it < 4; ++it) {
        const int li = r0 + it * 16;
        const int gi = ty0 + li;                  // global row
        const int gj0 = tx0 + c4;                 // first global col of group
        const long long n = base + (long long)gi * W_ + gj0;

        // Streaming inputs: one NT b128 load per array per 4 elements.
        const v4f u4 = __builtin_nontemporal_load((const v4f*)&U[n]);
        const v4f v4 = __builtin_nontemporal_load((const v4f*)&V[n]);
        const v4f p4 = __builtin_nontemporal_load((const v4f*)&img[n]);

        // Prefetch next iteration's rows (global_prefetch_b8).
        if (it < 3) {
            const long long np = n + 16LL * W_;
            __builtin_prefetch(&U[np],   0, 0);
            __builtin_prefetch(&V[np],   0, 0);
            __builtin_prefetch(&img[np], 0, 0);
        }

        #pragma unroll
        for (int e = 0; e < 4; ++e) {
            const int gj = gj0 + e;
            const float u = u4[e];
            const float v = v4[e];
            const float p = p4[e];

            // Exactly the reference math, fp32.
            const float X = fminf(fmaxf((float)gj + u, 0.0f), (float)(W_ - 1));
            const float Y = fminf(fmaxf((float)gi + v, 0.0f), (float)(H_ - 1));
            const int x0 = (int)floorf(X);
            const int y0 = (int)floorf(Y);
            const int x1 = min(x0 + 1, W_ - 1);
            const int y1 = min(y0 + 1, H_ - 1);
            const float fx = X - (float)x0;       // X_new - x0f
            const float fy = Y - (float)y0;       // Y_new - y0f
            const float gx = (float)x1 - X;       // x1f - X_new
            const float gy = (float)y1 - Y;       // y1f - Y_new
            const float a00 = p * (fx * fy);      // -> (y0, x0)
            const float a01 = p * (fx * gy);      // -> (y1, x0)
            const float a10 = p * (gx * fy);      // -> (y0, x1)
            const float a11 = p * (gx * gy);      // -> (y1, x1)

            // Local (halo-shifted) coordinates.
            const int lx0 = x0 - tx0 + HALO;
            const int ly0 = y0 - ty0 + HALO;
            const int lx1 = x1 - tx0 + HALO;      // lx0 or lx0+1
            const int ly1 = y1 - ty0 + HALO;

            if (lx0 >= 0 && ly0 >= 0 && lx1 < LTW && ly1 < LTW) {
                // Fast path: ds_add_f32 into the LDS tile.
                atomicAdd(&tile[ly0 * LSTRIDE + lx0], a00);
                atomicAdd(&tile[ly1 * LSTRIDE + lx0], a01);
                atomicAdd(&tile[ly0 * LSTRIDE + lx1], a10);
                atomicAdd(&tile[ly1 * LSTRIDE + lx1], a11);
            } else {
                // ~1e-14 probability fallback: direct global atomics.
                atomicAdd(&out[base + (long long)y0 * W_ + x0], a00);
                atomicAdd(&out[base + (long long)y1 * W_ + x0], a01);
                atomicAdd(&out[base + (long long)y0 * W_ + x1], a10);
                atomicAdd(&out[base + (long long)y1 * W_ + x1], a11);
            }
        }
    }
    __syncthreads();

    // Flush: one global_atomic_add_f32 per non-zero LDS cell.
    for (int i = threadIdx.x; i < LTW * LTW; i += NTHREADS) {
        const int ly = i / LTW;
        const int lx = i - ly * LTW;
        const float val = tile[ly * LSTRIDE + lx];
        if (val != 0.0f) {
            const int gy2 = ty0 + ly - HALO;
            const int gx2 = tx0 + lx - HALO;
            if (gy2 >= 0 && gy2 < H_ && gx2 >= 0 && gx2 < W_) {
                atomicAdd(&out[base + (long long)gy2 * W_ + gx2], val);
            }
        }
    }
}

extern "C" void kernel_launch(void* const* d_in, const int* in_sizes, int n_in,
                              void* d_out, int out_size, void* d_ws, size_t ws_size,
                              hipStream_t stream) {
    (void)in_sizes; (void)n_in; (void)d_ws; (void)ws_size; (void)out_size;

    const float* img = (const float*)d_in[0];   // input_image
    const float* U   = (const float*)d_in[1];   // U
    const float* V   = (const float*)d_in[2];   // V
    float* out       = (float*)d_out;

    const int N  = NPLANES * PLANE;             // 25,165,824 (== out_size)
    const int n4 = N / 4;

    // 1) Deterministic zero-init of the output accumulator.
    zero_f4<<<(n4 + 255) / 256, 256, 0, stream>>>((v4f*)out, n4);

    // 2) Tiled scatter: 16x16 tiles x 24 planes = 6144 workgroups.
    const int nblocks = (W_ / TILE) * (H_ / TILE) * NPLANES;
    advect_scatter<<<nblocks, NTHREADS, 0, stream>>>(img, U, V, out);
}